// GCN_2_3246995276080
// MI455X (gfx1250) — compile-verified
//
#include <hip/hip_runtime.h>

// ---------------------------------------------------------------------------
// GCN 2-layer for MI455X (gfx1250, wave32).
// Memory-bound workload (~1.6 GB edge traffic -> ~70us at 23.3 TB/s).
// Dense transforms use V_WMMA_F32_16X16X4_F32 (full fp32 precision).
// Layer 1 is transform-first (aggregate(H@W1)) to halve edge traffic.
// ---------------------------------------------------------------------------

typedef __attribute__((ext_vector_type(2))) float v2f;
typedef __attribute__((ext_vector_type(8))) float v8f;

#define N_NODES 50000
#define N_EDGES 800000
#define IN_DIM  128
#define HID_DIM 128
#define OUT_DIM 64

// ---------------- utility fills ----------------
__global__ void fill_kernel(float* __restrict__ p, float v, int n) {
    int i = blockIdx.x * blockDim.x + threadIdx.x;
    if (i < n) p[i] = v;
}

// deg[dst] += 1 for every edge (deg pre-filled with 1.0 for self-loops)
__global__ void deg_accum_kernel(const int* __restrict__ dst,
                                 float* __restrict__ deg, int nE) {
    int e = blockIdx.x * blockDim.x + threadIdx.x;
    if (e < nE) atomicAdd(&deg[dst[e]], 1.0f);
}

__global__ void dinv_kernel(float* __restrict__ d, int n) {
    int i = blockIdx.x * blockDim.x + threadIdx.x;
    if (i < n) d[i] = rsqrtf(fmaxf(d[i], 1.0f));
}

// ---------------- fp32 WMMA GEMM: C[M x NCOL] = A[M x 128] * W[128 x NCOL] --
// One 16x16 output tile per wave32, K stepped by 4 via V_WMMA_F32_16X16X4_F32.
// A frag (16x4 f32, ISA 7.12.2): lanes 0-15 -> {K=0,K=1}, lanes 16-31 -> {K=2,K=3}
// B frag (4x16 f32): VGPR0/1 hold rows {koff, koff+1}, N striped across lane&15
// C/D frag: VGPR r -> (M = r + 8*(lane>=16), N = lane&15)
template<int NCOL>
__global__ __launch_bounds__(256)
void gemm_wmma_f32(const float* __restrict__ A, const float* __restrict__ W,
                   float* __restrict__ C, int M) {
    const int lane   = threadIdx.x & 31;
    const int wave   = threadIdx.x >> 5;
    const int tile_m = blockIdx.x * 8 + wave;
    if (tile_m >= (M >> 4)) return;
    const int tile_n = blockIdx.y;

    const int half = lane >> 4;      // 0: lanes 0-15, 1: lanes 16-31
    const int lid  = lane & 15;
    const int row  = tile_m * 16 + lid;      // A row this lane supplies
    const int col  = tile_n * 16 + lid;      // B/C column this lane supplies
    const int koff = half * 2;               // K sub-offset per lane half

    const float* __restrict__ arow = A + (size_t)row * IN_DIM;

    v8f acc = {};
#pragma unroll 8
    for (int k = 0; k < IN_DIM; k += 4) {
        // 8-byte aligned pair load: A[row][k+koff], A[row][k+koff+1]
        v2f a = *(const v2f*)(arow + k + koff);
        v2f b;
        b.x = W[(size_t)(k + koff)     * NCOL + col];
        b.y = W[(size_t)(k + koff + 1) * NCOL + col];
        // (neg_a, A, neg_b, B, c_mod, C, reuse_a, reuse_b)
        acc = __builtin_amdgcn_wmma_f32_16x16x4_f32(
            false, a, false, b, (short)0, acc, false, false);
    }

#pragma unroll
    for (int r = 0; r < 8; ++r) {
        C[(size_t)(tile_m * 16 + r + half * 8) * NCOL + col] = acc[r];
    }
}

// ---------------- edge scatter-add: agg[dst] += Y[src] * dinv[s]*dinv[d] ----
// One wave32 per edge; each lane owns D/32 contiguous floats (coalesced
// global_load_b128/b64 of the source row, then fp32 global atomics).
template<int D>
__global__ __launch_bounds__(256)
void edge_agg_kernel(const int* __restrict__ src, const int* __restrict__ dst,
                     const float* __restrict__ Y, const float* __restrict__ dinv,
                     float* __restrict__ agg, int nE) {
    const int lane = threadIdx.x & 31;
    const int e    = blockIdx.x * 8 + (threadIdx.x >> 5);
    if (e >= nE) return;

    const int   s    = src[e];
    const int   d    = dst[e];
    const float norm = dinv[s] * dinv[d];

    constexpr int VEC = D / 32;  // 4 (layer0) or 2 (layer1)
    const float* __restrict__ yrow = Y   + (size_t)s * D + lane * VEC;
    float*       __restrict__ arow = agg + (size_t)d * D + lane * VEC;

    float v[VEC];
#pragma unroll
    for (int j = 0; j < VEC; ++j) v[j] = yrow[j];   // merges into b128/b64 load
#pragma unroll
    for (int j = 0; j < VEC; ++j) atomicAdd(&arow[j], v[j] * norm);
}

// ---------------- self-loop + bias (+ReLU): acc += Y*dinv^2 + b ------------
template<int D, bool RELU>
__global__ __launch_bounds__(256)
void self_bias_kernel(float* __restrict__ acc, const float* __restrict__ Y,
                      const float* __restrict__ dinv, const float* __restrict__ bias,
                      int n) {
    int i = blockIdx.x * blockDim.x + threadIdx.x;
    if (i >= n * D) return;
    const int   node = i / D;
    const int   c    = i % D;
    const float di   = dinv[node];
    float v = acc[i] + Y[i] * di * di + bias[c];
    acc[i] = RELU ? fmaxf(v, 0.0f) : v;
}

// ---------------------------------------------------------------------------
extern "C" void kernel_launch(void* const* d_in, const int* in_sizes, int n_in,
                              void* d_out, int out_size, void* d_ws, size_t ws_size,
                              hipStream_t stream) {
    // setup_inputs order: V, E, X, W0, b0, W1, b1
    const int*   E  = (const int*)  d_in[1];   // [2, N_EDGES]: row0=src, row1=dst
    const float* X  = (const float*)d_in[2];   // [N, 128]
    const float* W0 = (const float*)d_in[3];   // [128, 128]
    const float* b0 = (const float*)d_in[4];   // [128]
    const float* W1 = (const float*)d_in[5];   // [128, 64]
    const float* b1 = (const float*)d_in[6];   // [64]
    float* out = (float*)d_out;                // [N, 64]

    const int* Esrc = E;
    const int* Edst = E + N_EDGES;

    // workspace layout (floats): dinv | agg0/H | xw (XW0, later Y1)
    float* ws   = (float*)d_ws;
    float* dinv = ws;                                   // 50000 (pad to 50048)
    float* agg0 = ws + 50048;                           // N*128
    float* xw   = agg0 + (size_t)N_NODES * HID_DIM;     // N*128 (reused as N*64)

    const int T = 256;
    // --- per-call (graph-replay-safe) init ---
    fill_kernel<<<(N_NODES + T - 1) / T, T, 0, stream>>>(dinv, 1.0f, N_NODES); // self-loop deg
    fill_kernel<<<(N_NODES * HID_DIM + T - 1) / T, T, 0, stream>>>(agg0, 0.0f, N_NODES * HID_DIM);
    fill_kernel<<<(N_NODES * OUT_DIM + T - 1) / T, T, 0, stream>>>(out, 0.0f, N_NODES * OUT_DIM);

    // --- degrees -> dinv ---
    deg_accum_kernel<<<(N_EDGES + T - 1) / T, T, 0, stream>>>(Edst, dinv, N_EDGES);
    dinv_kernel<<<(N_NODES + T - 1) / T, T, 0, stream>>>(dinv, N_NODES);

    const int tilesM = N_NODES / 16;  // 3125 exactly

    // --- layer 0: XW0 = X@W0 ; agg ; H = relu(agg + self + b0) ---
    dim3 g0((tilesM + 7) / 8, HID_DIM / 16);
    gemm_wmma_f32<HID_DIM><<<g0, 256, 0, stream>>>(X, W0, xw, N_NODES);
    edge_agg_kernel<HID_DIM><<<(N_EDGES + 7) / 8, 256, 0, stream>>>(Esrc, Edst, xw, dinv, agg0, N_EDGES);
    self_bias_kernel<HID_DIM, true><<<(N_NODES * HID_DIM + T - 1) / T, T, 0, stream>>>(agg0, xw, dinv, b0, N_NODES);

    // --- layer 1 (transform-first): Y1 = H@W1 ; out = aggregate(Y1) + b1 ---
    dim3 g1((tilesM + 7) / 8, OUT_DIM / 16);
    gemm_wmma_f32<OUT_DIM><<<g1, 256, 0, stream>>>(agg0, W1, xw, N_NODES);
    edge_agg_kernel<OUT_DIM><<<(N_EDGES + 7) / 8, 256, 0, stream>>>(Esrc, Edst, xw, dinv, out, N_EDGES);
    self_bias_kernel<OUT_DIM, false><<<(N_NODES * OUT_DIM + T - 1) / T, T, 0, stream>>>(out, xw, dinv, b1, N_NODES);
}